// GraphSAGE_5617817223572
// MI455X (gfx1250) — compile-verified
//
#include <hip/hip_runtime.h>
#include <hip/hip_bf16.h>
#include <math.h>

#define N_NODES    100000
#define N_EDGES    1600000
#define DIM_IN     64
#define DIM_H      128
#define DIM_OUT    2
#define NUM_GRAPHS 64

typedef __attribute__((ext_vector_type(2))) float v2f;
typedef __attribute__((ext_vector_type(8))) float v8f;

// ---------------------------------------------------------------------------
// 1) in-degree via f32 atomics
__global__ void deg_kernel(const int* __restrict__ dst, float* __restrict__ deg) {
    int e = blockIdx.x * blockDim.x + threadIdx.x;
    if (e < N_EDGES) atomicAdd(&deg[dst[e]], 1.0f);
}

// deg -> 1/max(deg,1) in place
__global__ void invdeg_kernel(float* __restrict__ deg) {
    int i = blockIdx.x * blockDim.x + threadIdx.x;
    if (i < N_NODES) deg[i] = 1.0f / fmaxf(deg[i], 1.0f);
}

// ---------------------------------------------------------------------------
// 2) scatter-add x[src] -> agg1[dst]   (DIM_IN=64 floats = 16 float4 per edge)
__global__ void scatter1_kernel(const float* __restrict__ x,
                                const int* __restrict__ src,
                                const int* __restrict__ dst,
                                float* __restrict__ agg1) {
    unsigned tid = blockIdx.x * blockDim.x + threadIdx.x;   // E*16 = 25.6M
    if (tid >= (unsigned)N_EDGES * 16u) return;
    int e = tid >> 4, q = tid & 15;
    int s = src[e], d = dst[e];
    float4 v = ((const float4*)x)[(size_t)s * 16 + q];
    float* base = &agg1[(size_t)d * DIM_IN + q * 4];
    atomicAdd(base + 0, v.x); atomicAdd(base + 1, v.y);
    atomicAdd(base + 2, v.z); atomicAdd(base + 3, v.w);
}

// ---------------------------------------------------------------------------
// 3) layer 1: h1 = relu((agg1*invdeg) @ W1l + x @ W1r + b1), full f32 WMMA.
//    Block = 256 threads = 8 waves; wave w owns column tile [16w, 16w+16),
//    block b owns node rows [16b, 16b+16).  K=64 -> 16 x (16x16x4) steps,
//    two WMMAs per step (neighbor term + root term) accumulating in C.
__global__ void __launch_bounds__(256)
layer1_wmma_kernel(const float* __restrict__ x,
                   const float* __restrict__ agg1,
                   const float* __restrict__ invdeg,
                   const float* __restrict__ W1l,
                   const float* __restrict__ W1r,
                   const float* __restrict__ b1,
                   float* __restrict__ h1) {
    const int lane = threadIdx.x & 31;
    const int wave = threadIdx.x >> 5;         // 0..7  -> column tile
    const int half = lane >> 4;                // lane half selects K pair
    const int n16  = lane & 15;
    const int nodebase = blockIdx.x * 16;      // N_NODES = 6250 * 16 exactly
    const int colbase  = wave * 16;
    const int row = nodebase + n16;            // A-fragment row (M = lane&15)
    const int col = colbase + n16;             // B/C-fragment column

    const float s = invdeg[row];

    // C init = bias (same value down every row of the column)
    v8f c;
    const float bias = b1[col];
#pragma unroll
    for (int v = 0; v < 8; ++v) c[v] = bias;

    const float* aggRow = agg1 + (size_t)row * DIM_IN;
    const float* xRow   = x    + (size_t)row * DIM_IN;

#pragma unroll
    for (int k = 0; k < DIM_IN; k += 4) {
        const int kb = k + half * 2;           // lanes 0-15: K=k,k+1 ; 16-31: K=k+2,k+3
        v2f aA, aX, bL, bR;
        aA.x = aggRow[kb]     * s;  aA.y = aggRow[kb + 1] * s;
        aX.x = xRow[kb];            aX.y = xRow[kb + 1];
        bL.x = W1l[kb * DIM_H + col];  bL.y = W1l[(kb + 1) * DIM_H + col];
        bR.x = W1r[kb * DIM_H + col];  bR.y = W1r[(kb + 1) * DIM_H + col];
        c = __builtin_amdgcn_wmma_f32_16x16x4_f32(false, aA, false, bL,
                                                  (short)0, c, false, false);
        c = __builtin_amdgcn_wmma_f32_16x16x4_f32(false, aX, false, bR,
                                                  (short)0, c, false, false);
    }

    // ReLU + store: C/D layout -> M = v + 8*half, N = lane&15
#pragma unroll
    for (int v = 0; v < 8; ++v) {
        int m = v + half * 8;
        h1[(size_t)(nodebase + m) * DIM_H + col] = fmaxf(c[v], 0.0f);
    }
}

// ---------------------------------------------------------------------------
// 4) scatter-add h1[src] -> agg2[dst]  (DIM_H=128 floats = 32 float4 per edge)
__global__ void scatter2_kernel(const float* __restrict__ h1,
                                const int* __restrict__ src,
                                const int* __restrict__ dst,
                                float* __restrict__ agg2) {
    unsigned tid = blockIdx.x * blockDim.x + threadIdx.x;   // E*32 = 51.2M
    if (tid >= (unsigned)N_EDGES * 32u) return;
    int e = tid >> 5, q = tid & 31;
    int s = src[e], d = dst[e];
    float4 v = ((const float4*)h1)[(size_t)s * 32 + q];
    float* base = &agg2[(size_t)d * DIM_H + q * 4];
    atomicAdd(base + 0, v.x); atomicAdd(base + 1, v.y);
    atomicAdd(base + 2, v.z); atomicAdd(base + 3, v.w);
}

// ---------------------------------------------------------------------------
// 5) layer 2 (128 -> 2, VALU dot with W2 in LDS) fused with graph mean-pool
//    accumulation via atomics.
__global__ void __launch_bounds__(256)
layer2_pool_kernel(const float* __restrict__ h1,
                   const float* __restrict__ agg2,
                   const float* __restrict__ invdeg,
                   const float* __restrict__ W2l,
                   const float* __restrict__ W2r,
                   const float* __restrict__ b2,
                   const int* __restrict__ batch,
                   float* __restrict__ pool,
                   float* __restrict__ cnt) {
    __shared__ float sWl[DIM_H * DIM_OUT];
    __shared__ float sWr[DIM_H * DIM_OUT];
    if (threadIdx.x < DIM_H * DIM_OUT) {
        sWl[threadIdx.x] = W2l[threadIdx.x];
        sWr[threadIdx.x] = W2r[threadIdx.x];
    }
    __syncthreads();

    int n = blockIdx.x * blockDim.x + threadIdx.x;
    if (n >= N_NODES) return;

    const float s = invdeg[n];
    float acc0 = b2[0], acc1 = b2[1];
    const float* aRow = agg2 + (size_t)n * DIM_H;
    const float* hRow = h1   + (size_t)n * DIM_H;
#pragma unroll 8
    for (int d = 0; d < DIM_H; ++d) {
        float a = aRow[d] * s;
        float h = hRow[d];
        acc0 += a * sWl[d * 2 + 0] + h * sWr[d * 2 + 0];
        acc1 += a * sWl[d * 2 + 1] + h * sWr[d * 2 + 1];
    }
    int g = batch[n];
    atomicAdd(&pool[g * 2 + 0], acc0);
    atomicAdd(&pool[g * 2 + 1], acc1);
    atomicAdd(&cnt[g], 1.0f);
}

// ---------------------------------------------------------------------------
// 6) finalize: mean, log-softmax over 2 classes, write both outputs
__global__ void finalize_kernel(const float* __restrict__ pool,
                                const float* __restrict__ cnt,
                                float* __restrict__ out) {
    int g = threadIdx.x;
    if (g >= NUM_GRAPHS) return;
    float c  = fmaxf(cnt[g], 1.0f);
    float h0 = pool[g * 2 + 0] / c;
    float h1 = pool[g * 2 + 1] / c;
    float m   = fmaxf(h0, h1);
    float lse = m + logf(expf(h0 - m) + expf(h1 - m));
    out[g * 2 + 0] = h0;
    out[g * 2 + 1] = h1;
    out[NUM_GRAPHS * DIM_OUT + g * 2 + 0] = h0 - lse;
    out[NUM_GRAPHS * DIM_OUT + g * 2 + 1] = h1 - lse;
}

// ---------------------------------------------------------------------------
extern "C" void kernel_launch(void* const* d_in, const int* in_sizes, int n_in,
                              void* d_out, int out_size, void* d_ws, size_t ws_size,
                              hipStream_t stream) {
    const float* x    = (const float*)d_in[0];
    const int*   edge = (const int*)d_in[1];
    const int*   bat  = (const int*)d_in[2];
    const float* W1l  = (const float*)d_in[3];
    const float* W1r  = (const float*)d_in[4];
    const float* b1   = (const float*)d_in[5];
    const float* W2l  = (const float*)d_in[6];
    const float* W2r  = (const float*)d_in[7];
    const float* b2   = (const float*)d_in[8];
    const int* src = edge;             // edge_index row 0
    const int* dst = edge + N_EDGES;   // edge_index row 1

    char* ws = (char*)d_ws;
    size_t off = 0;
    auto carve = [&](size_t bytes) {
        void* p = ws + off;
        off = (off + bytes + 255) & ~(size_t)255;
        return p;
    };
    float* deg  = (float*)carve((size_t)N_NODES * 4);             // becomes invdeg
    float* agg1 = (float*)carve((size_t)N_NODES * DIM_IN * 4);    // 25.6 MB
    float* h1   = (float*)carve((size_t)N_NODES * DIM_H * 4);     // 51.2 MB
    float* agg2 = (float*)carve((size_t)N_NODES * DIM_H * 4);     // 51.2 MB
    float* pool = (float*)carve((size_t)NUM_GRAPHS * DIM_OUT * 4);
    float* cnt  = (float*)carve((size_t)NUM_GRAPHS * 4);

    // zero accumulators every call (graph-capture safe memset nodes)
    hipMemsetAsync(deg,  0, (size_t)N_NODES * 4, stream);
    hipMemsetAsync(agg1, 0, (size_t)N_NODES * DIM_IN * 4, stream);
    hipMemsetAsync(agg2, 0, (size_t)N_NODES * DIM_H * 4, stream);
    hipMemsetAsync(pool, 0, (size_t)NUM_GRAPHS * DIM_OUT * 4, stream);
    hipMemsetAsync(cnt,  0, (size_t)NUM_GRAPHS * 4, stream);

    deg_kernel<<<(N_EDGES + 255) / 256, 256, 0, stream>>>(dst, deg);
    invdeg_kernel<<<(N_NODES + 255) / 256, 256, 0, stream>>>(deg);

    scatter1_kernel<<<(N_EDGES * 16u + 255) / 256, 256, 0, stream>>>(x, src, dst, agg1);

    layer1_wmma_kernel<<<N_NODES / 16, 256, 0, stream>>>(x, agg1, deg, W1l, W1r, b1, h1);

    scatter2_kernel<<<((unsigned)N_EDGES * 32u + 255) / 256, 256, 0, stream>>>(h1, src, dst, agg2);

    layer2_pool_kernel<<<(N_NODES + 255) / 256, 256, 0, stream>>>(
        h1, agg2, deg, W2l, W2r, b2, bat, pool, cnt);

    finalize_kernel<<<1, 64, 0, stream>>>(pool, cnt, (float*)d_out);
}